// FlowMambaCell_54820962566287
// MI455X (gfx1250) — compile-verified
//
#include <hip/hip_runtime.h>
#include <math.h>

typedef __attribute__((ext_vector_type(2))) float v2f;
typedef __attribute__((ext_vector_type(8))) float v8f;

// Sizes (fixed by the reference)
#define BATCH   4
#define DMODEL  64
#define DSTATE  16
#define HH      32
#define WW      32
#define NV      9
#define HW      (HH * WW)            // 1024
#define K_RED   (DMODEL * 9)         // 576
#define NPIX    (BATCH * HW)         // 4096
#define NCH_OUT 96                   // 64 delta + 16 B + 16 C
#define PIX_TILES (NPIX / 16)        // 256
#define CH_TILES  (NCH_OUT / 16)     // 6
#define NTILES   (PIX_TILES * CH_TILES) // 1536

#define Y_OUT_ELEMS  (BATCH * NV * DMODEL * HW)            // 2,359,296
#define S_ELEMS      (BATCH * NV * DMODEL * DSTATE * HW)   // 37,748,736

// ---------------------------------------------------------------------------
// Kernel 1: fused 3x3 circular convs (delta_raw, B_val, C_val) as implicit
// GEMM with V_WMMA_F32_16X16X4_F32.  M=4096 pixels, N=96 out ch, K=576.
// One wave computes one 16x16 output tile.
// ---------------------------------------------------------------------------
__global__ __launch_bounds__(256) void
conv_wmma_kernel(const float* __restrict__ u,
                 const float* __restrict__ Wd,   // (64,64,3,3)
                 const float* __restrict__ Wb,   // (16,64,3,3)
                 const float* __restrict__ Wc,   // (16,64,3,3)
                 float* __restrict__ ws_delta,   // (4,64,32,32)
                 float* __restrict__ ws_B,       // (4,16,32,32)
                 float* __restrict__ ws_C)       // (4,16,32,32)
{
    const int wave = threadIdx.x >> 5;
    const int lane = threadIdx.x & 31;
    const int tile = blockIdx.x * 8 + wave;      // 0..1535
    if (tile >= NTILES) return;

    const int ct = tile % CH_TILES;              // channel tile 0..5
    const int pt = tile / CH_TILES;              // pixel tile   0..255
    const int p0 = pt << 4;                      // first of 16 pixels
    const int b  = p0 >> 10;                     // /1024
    const int y  = (p0 >> 5) & 31;
    const int x0 = p0 & 31;                      // 0 or 16 (row-aligned tiles)

    const int m  = lane & 15;                    // A: pixel row M; B: out ch N
    const int hi = lane >> 4;                    // K half select

    // Per-lane weight column (out channel N = m)
    const int co = ct * 16 + m;
    const float* wbase;
    int corow;
    if (co < 64)      { wbase = Wd; corow = co;      }
    else if (co < 80) { wbase = Wb; corow = co - 64; }
    else              { wbase = Wc; corow = co - 80; }

    v8f acc = {};

    // K ordering: k = tap*64 + ci  (tap = 3*ky + kx).  4 | 64 so each K-chunk
    // stays inside one tap.
    #pragma unroll 4
    for (int k0 = 0; k0 < K_RED; k0 += 4) {
        const int tap = k0 >> 6;
        const int dy  = tap / 3 - 1;
        const int dx  = tap % 3 - 1;
        const int kk  = (k0 & 63) + hi * 2;      // ci for this lane's slot 0
        const int yy  = (y + dy) & 31;
        const int xx  = (x0 + m + dx) & 31;

        // A: u[b, ci, yy, xx]  (pixel M=m, K=kk / kk+1)
        const float* up = u + (size_t)b * DMODEL * HW + yy * WW + xx;
        v2f a;
        a.x = up[(kk    ) * HW];
        a.y = up[(kk + 1) * HW];

        // B: W[co, ci=kk, tap]  (row K=kk / kk+1, col N=m)
        const float* wp = wbase + ((size_t)corow * DMODEL + kk) * 9 + tap;
        v2f bm;
        bm.x = wp[0];
        bm.y = wp[9];

        acc = __builtin_amdgcn_wmma_f32_16x16x4_f32(
            /*neg_a=*/false, a, /*neg_b=*/false, bm,
            /*c_mod=*/(short)0, acc, /*reuse_a=*/false, /*reuse_b=*/false);
    }

    // D layout: VGPR r -> pixel M = r + 8*hi (=> x = x0 + r + 8*hi), N = m
    const int xb = x0 + hi * 8;
    #pragma unroll
    for (int r = 0; r < 8; ++r) {
        const float vr = acc[r];
        const int   x  = xb + r;
        const int   hwo = y * WW + x;
        if (ct < 4)
            ws_delta[((size_t)b * DMODEL + co) * HW + hwo] = vr;
        else if (ct == 4)
            ws_B[((size_t)b * DSTATE + corow) * HW + hwo] = vr;
        else
            ws_C[((size_t)b * DSTATE + corow) * HW + hwo] = vr;
    }
}

// ---------------------------------------------------------------------------
// Kernel 2: fused pointwise + state update + y reduction.
// One block per (b, v, d); 1024 threads = one (y,x) cell each.  Each wave owns
// one 32-float row, so the circular x-shift is a rotation within one 128B
// line -> fully coalesced streaming of s_prev/s_next (the 300 MB that set the
// ~14us roofline).  Small tensors (delta/B/C, 1.5 MB) hit L2 on reuse.
// ---------------------------------------------------------------------------
__global__ __launch_bounds__(1024) void
state_update_kernel(const float* __restrict__ u,
                    const float* __restrict__ s_prev,
                    const float* __restrict__ b_delta,
                    const float* __restrict__ log_A_real, // (64,16)
                    const float* __restrict__ Dvec,
                    const float* __restrict__ dt_scalar,
                    const float* __restrict__ ws_delta,
                    const float* __restrict__ ws_B,
                    const float* __restrict__ ws_C,
                    float* __restrict__ y_out,
                    float* __restrict__ s_next)
{
    const int bi = blockIdx.x;               // b*576 + v*64 + d
    const int d  = bi & 63;
    const int v  = (bi >> 6) % NV;
    const int b  = bi / (NV * DMODEL);

    const int t  = threadIdx.x;              // 0..1023
    const int y  = t >> 5;
    const int x  = t & 31;

    __shared__ float sA[DSTATE];
    if (t < DSTATE) sA[t] = -__expf(log_A_real[d * DSTATE + t]);
    __syncthreads();

    const float dt  = dt_scalar[0];
    const size_t bd = (size_t)(b * DMODEL + d) * HW + t;

    // delta = softplus(conv + b_delta + dt)
    const float pre   = ws_delta[bd] + b_delta[d] + dt;
    const float delta = (pre > 20.0f) ? pre : log1pf(__expf(pre));

    const float uu = u[bd];
    const float uD = uu * Dvec[d];

    // SHIFTS[v] = (vx, vy) with vx = v/3-1, vy = v%3-1; roll(-vy,-vx) =>
    // s_trans[y][x] = s_prev[(y+vy)%32][(x+vx)%32]
    const int vx = v / 3 - 1;
    const int vy = v % 3 - 1;
    const int yp = (y + vy) & 31;
    const int xp = (x + vx) & 31;

    const size_t sbase = (size_t)((b * NV + v) * DMODEL + d) * DSTATE * HW;
    const float* sp = s_prev + sbase + (size_t)yp * WW + xp;
    float*       sn = s_next + sbase + t;

    const size_t bcn = (size_t)b * DSTATE * HW + t;

    float yacc = 0.0f;
    #pragma unroll
    for (int n = 0; n < DSTATE; ++n) {
        const float An   = sA[n];
        const float abar = __expf(delta * An);
        const float Bv   = ws_B[bcn + (size_t)n * HW];
        const float bu   = (abar - 1.0f) / An * Bv * uu;
        const float s    = sp[(size_t)n * HW];
        const float snv  = fmaf(abar, s, bu);
        sn[(size_t)n * HW] = snv;
        const float Cv = ws_C[bcn + (size_t)n * HW];
        yacc = fmaf(snv, Cv, yacc);
    }

    y_out[(size_t)((b * NV + v) * DMODEL + d) * HW + t] = yacc + uD;
}

// ---------------------------------------------------------------------------
extern "C" void kernel_launch(void* const* d_in, const int* in_sizes, int n_in,
                              void* d_out, int out_size, void* d_ws, size_t ws_size,
                              hipStream_t stream) {
    const float* u_t        = (const float*)d_in[0];
    const float* s_prev     = (const float*)d_in[1];
    const float* W_delta    = (const float*)d_in[2];
    const float* b_delta    = (const float*)d_in[3];
    const float* W_B        = (const float*)d_in[4];
    const float* W_C        = (const float*)d_in[5];
    const float* log_A_real = (const float*)d_in[6];
    const float* Dvec       = (const float*)d_in[7];
    const float* dt_scalar  = (const float*)d_in[8];

    float* ws       = (float*)d_ws;
    float* ws_delta = ws;                                  // 262144
    float* ws_B     = ws_delta + BATCH * DMODEL * HW;      // 65536
    float* ws_C     = ws_B     + BATCH * DSTATE * HW;      // 65536

    float* y_out  = (float*)d_out;
    float* s_next = y_out + Y_OUT_ELEMS;

    // Conv stage: 1536 wave-tiles, 8 waves/block -> 192 blocks of 256.
    conv_wmma_kernel<<<NTILES / 8, 256, 0, stream>>>(
        u_t, W_delta, W_B, W_C, ws_delta, ws_B, ws_C);

    // State stage: one block per (b, v, d).
    state_update_kernel<<<BATCH * NV * DMODEL, 1024, 0, stream>>>(
        u_t, s_prev, b_delta, log_A_real, Dvec, dt_scalar,
        ws_delta, ws_B, ws_C, y_out, s_next);
}